// Net_79001628442700
// MI455X (gfx1250) — compile-verified
//
#include <hip/hip_runtime.h>

// y[i] = W2[i,:] . relu(x[i]*W1[i,:] + b1[i,:]) + b2[i],  H = 8
//
// Memory-bound (AI ~ 0.37 flop/byte): 432 MB moved, 160 MFLOP.
// Strategy: fully-coalesced 128-bit non-temporal streaming loads,
// grid-stride loop with L2 prefetch one stride ahead, NT store.

typedef float v4f __attribute__((ext_vector_type(4)));

__global__ __launch_bounds__(256) void tiny_mlp_stream_kernel(
    const float* __restrict__ x,
    const float* __restrict__ W1,
    const float* __restrict__ b1,
    const float* __restrict__ W2,
    const float* __restrict__ b2,
    float* __restrict__ y,
    int n)
{
    const int stride = gridDim.x * blockDim.x;
    for (int i = blockIdx.x * blockDim.x + threadIdx.x; i < n; i += stride) {
        // Prefetch next grid-stride iteration's lines into L2 while this
        // iteration's loads are in flight (gfx1250 global_prefetch_b8).
        const int ip = i + stride;
        if (ip < n) {
            __builtin_prefetch(x  + ip,     0, 1);
            __builtin_prefetch(W1 + ip * 8, 0, 1);
            __builtin_prefetch(b1 + ip * 8, 0, 1);
            __builtin_prefetch(W2 + ip * 8, 0, 1);
            __builtin_prefetch(b2 + ip,     0, 1);
        }

        // Streamed once, working set (432 MB) >> L2 (192 MB) -> non-temporal.
        const float xi  = __builtin_nontemporal_load(x + i);
        const float b2i = __builtin_nontemporal_load(b2 + i);

        const v4f* W1v = (const v4f*)(W1 + (size_t)i * 8);
        const v4f* b1v = (const v4f*)(b1 + (size_t)i * 8);
        const v4f* W2v = (const v4f*)(W2 + (size_t)i * 8);

        const v4f w1a = __builtin_nontemporal_load(W1v + 0);
        const v4f w1b = __builtin_nontemporal_load(W1v + 1);
        const v4f b1a = __builtin_nontemporal_load(b1v + 0);
        const v4f b1b = __builtin_nontemporal_load(b1v + 1);
        const v4f w2a = __builtin_nontemporal_load(W2v + 0);
        const v4f w2b = __builtin_nontemporal_load(W2v + 1);

        float acc = b2i;
#pragma unroll
        for (int k = 0; k < 4; ++k) {
            const float ha = fmaxf(fmaf(xi, w1a[k], b1a[k]), 0.0f);
            const float hb = fmaxf(fmaf(xi, w1b[k], b1b[k]), 0.0f);
            acc = fmaf(ha, w2a[k], acc);
            acc = fmaf(hb, w2b[k], acc);
        }

        __builtin_nontemporal_store(acc, y + i);
    }
}

extern "C" void kernel_launch(void* const* d_in, const int* in_sizes, int n_in,
                              void* d_out, int out_size, void* d_ws, size_t ws_size,
                              hipStream_t stream)
{
    const float* x  = (const float*)d_in[0];
    const float* W1 = (const float*)d_in[1];
    const float* b1 = (const float*)d_in[2];
    const float* W2 = (const float*)d_in[3];
    const float* b2 = (const float*)d_in[4];
    float* y = (float*)d_out;

    const int n = in_sizes[0];          // N (x is [N,1])

    const int threads = 256;            // 8 wave32 waves per workgroup
    int blocks = 4096;                  // ~1M threads, ~4 grid-stride iters
    const int needed = (n + threads - 1) / threads;
    if (blocks > needed) blocks = needed;

    tiny_mlp_stream_kernel<<<blocks, threads, 0, stream>>>(x, W1, b1, W2, b2, y, n);
}